// S9SSMKernel_28535762715090
// MI455X (gfx1250) — compile-verified
//
#include <hip/hip_runtime.h>
#include <hip/hip_bf16.h>

// S4D SSM kernel materialization via chunked-scan + FP32 WMMA on gfx1250.
//
//   K[d, 16q+r] = sum_n M[q,n] * P[n,r]
//   M[q,n] = (C*B)[d,n] * (A_bar^16)^q    (128 x 64 complex, per channel)
//   P[n,r] = A_bar[d,n]^r                 (64 x 16  complex, per channel)
//
// Complex GEMM -> 4 real GEMMs -> chains of V_WMMA_F32_16X16X4_F32.

#define D_MODEL 512
#define N_STATE 64
#define L_SEQ   2048

typedef __attribute__((ext_vector_type(2))) float v2f;
typedef __attribute__((ext_vector_type(8))) float v8f;

__global__ __launch_bounds__(128) void s4d_wmma_kernel(
    const float* __restrict__ log_A_real,  // (512, 64)
    const float* __restrict__ A_imag,      // (64,)
    const float* __restrict__ B_re,        // (512, 64)
    const float* __restrict__ B_im,        // (512, 64)
    const float* __restrict__ C_re,        // (512, 64)
    const float* __restrict__ C_im,        // (512, 64)
    const float* __restrict__ log_dt,      // (512,)
    float2* __restrict__ out)              // (512, 2048) complex64 interleaved
{
    // LDS: M half-buffer (64 q-rows), P, and per-n constants.
    // Row padding (65 / 17 floats) avoids 16-way LDS bank conflicts on the
    // strided A-fragment reads (64 banks x 4B => 256B period).
    __shared__ float sMre[64][65];
    __shared__ float sMim[64][65];
    __shared__ float sPre[64][17];
    __shared__ float sPim[64][17];
    __shared__ float sPimN[64][17];
    __shared__ float sA16re[N_STATE], sA16im[N_STATE];

    const int d    = blockIdx.x;          // channel
    const int tid  = threadIdx.x;         // 0..127
    const int wave = tid >> 5;            // 0..3 (wave32)
    const int lane = tid & 31;
    const int row  = lane & 15;           // 0..15
    const int hi   = lane >> 4;           // 0 or 1 (K sub-pair select)

    // ---------------- per-(d,n) setup: threads 0..63, n = tid -------------
    float mre = 0.f, mim = 0.f;           // running M scan state (thread n)
    float a16re = 0.f, a16im = 0.f;
    if (tid < N_STATE) {
        const int n = tid;
        const float dtv = __expf(log_dt[d]);
        const float are = -__expf(log_A_real[d * N_STATE + n]) * dtv;
        const float th  = A_imag[n] * dtv;
        const float rad = __expf(are);
        const float abre = rad * __cosf(th);       // A_bar
        const float abim = rad * __sinf(th);

        const float bre = B_re[d * N_STATE + n], bim = B_im[d * N_STATE + n];
        const float cre = C_re[d * N_STATE + n], cim = C_im[d * N_STATE + n];
        mre = cre * bre - cim * bim;               // CB = C*B
        mim = cre * bim + cim * bre;

        // A_bar^16 by 4 complex squarings
        float xre = abre, xim = abim;
        #pragma unroll
        for (int s = 0; s < 4; ++s) {
            const float t = xre * xre - xim * xim;
            xim = 2.f * xre * xim;
            xre = t;
        }
        a16re = xre; a16im = xim;
        sA16re[n] = xre; sA16im[n] = xim;

        // P[n, r] = A_bar^r, r = 0..15  (and negated imag copy for c_re acc,
        // since f32 WMMA forbids A/B NEG modifiers)
        float pre = 1.f, pim = 0.f;
        #pragma unroll
        for (int r = 0; r < 16; ++r) {
            sPre[n][r]  = pre;
            sPim[n][r]  = pim;
            sPimN[n][r] = -pim;
            const float t = pre * abre - pim * abim;
            pim = pre * abim + pim * abre;
            pre = t;
        }
    }
    __syncthreads();

    // ---------------- two halves of 64 q-rows each -------------------------
    #pragma unroll 1
    for (int h = 0; h < 2; ++h) {
        // Fill sM[0..63][n] with M[h*64 + qq, n]; scan continues in registers.
        if (tid < N_STATE) {
            const int n = tid;
            float lre = mre, lim = mim;
            #pragma unroll 4
            for (int qq = 0; qq < 64; ++qq) {
                sMre[qq][n] = lre;
                sMim[qq][n] = lim;
                const float t = lre * a16re - lim * a16im;
                lim = lre * a16im + lim * a16re;
                lre = t;
            }
            mre = lre; mim = lim;                  // carry into next half
            (void)sA16re; (void)sA16im;
        }
        __syncthreads();

        // Each wave owns one 16-row q-tile of this half.
        const int qbase = wave * 16;               // within half
        v8f c_re = {0.f, 0.f, 0.f, 0.f, 0.f, 0.f, 0.f, 0.f};
        v8f c_im = {0.f, 0.f, 0.f, 0.f, 0.f, 0.f, 0.f, 0.f};

        #pragma unroll 4
        for (int kc = 0; kc < 16; ++kc) {
            const int k0 = kc * 4 + hi * 2;        // this lane-group's K pair

            // A fragments (16x4 f32): lanes 0-15 -> K={k0,k0+1} of rows 0..15,
            // lanes 16-31 -> K={k0,k0+1} with hi-offset, per ISA layout.
            v2f aRe, aIm;
            aRe.x = sMre[qbase + row][k0];
            aRe.y = sMre[qbase + row][k0 + 1];
            aIm.x = sMim[qbase + row][k0];
            aIm.y = sMim[qbase + row][k0 + 1];

            // B fragments (4x16 f32), mirrored lane layout: element (k, n=row).
            v2f bRe, bIm, bImN;
            bRe.x  = sPre [k0][row];  bRe.y  = sPre [k0 + 1][row];
            bIm.x  = sPim [k0][row];  bIm.y  = sPim [k0 + 1][row];
            bImN.x = sPimN[k0][row];  bImN.y = sPimN[k0 + 1][row];

            // Complex MAC: c_re += Mre*Pre + Mim*(-Pim); c_im += Mre*Pim + Mim*Pre
            c_re = __builtin_amdgcn_wmma_f32_16x16x4_f32(
                false, aRe, false, bRe, (short)0, c_re, false, false);
            c_re = __builtin_amdgcn_wmma_f32_16x16x4_f32(
                false, aIm, false, bImN, (short)0, c_re, false, false);
            c_im = __builtin_amdgcn_wmma_f32_16x16x4_f32(
                false, aRe, false, bIm, (short)0, c_im, false, false);
            c_im = __builtin_amdgcn_wmma_f32_16x16x4_f32(
                false, aIm, false, bRe, (short)0, c_im, false, false);
        }

        // Store: C/D layout — VGPR j holds row j (lanes 0-15) / row j+8
        // (lanes 16-31), column = lane&15.  l = q*16 + r, complex interleaved.
        const int qrow0 = h * 64 + qbase + hi * 8;
        float2* outd = out + (size_t)d * L_SEQ;
        #pragma unroll
        for (int j = 0; j < 8; ++j) {
            const int l = (qrow0 + j) * 16 + row;
            outd[l] = make_float2(c_re[j], c_im[j]);
        }
        __syncthreads();   // protect sM before next half overwrites it
    }
}

extern "C" void kernel_launch(void* const* d_in, const int* in_sizes, int n_in,
                              void* d_out, int out_size, void* d_ws, size_t ws_size,
                              hipStream_t stream) {
    // setup_inputs order:
    // 0: length (int scalar), 1: log_A_real, 2: A_imag, 3: B_re, 4: B_im,
    // 5: C_re, 6: C_im, 7: log_dt
    const float* log_A_real = (const float*)d_in[1];
    const float* A_imag     = (const float*)d_in[2];
    const float* B_re       = (const float*)d_in[3];
    const float* B_im       = (const float*)d_in[4];
    const float* C_re       = (const float*)d_in[5];
    const float* C_im       = (const float*)d_in[6];
    const float* log_dt     = (const float*)d_in[7];
    float2* out             = (float2*)d_out;

    dim3 grid(D_MODEL);   // one workgroup per channel
    dim3 block(128);      // 4 wave32s
    s4d_wmma_kernel<<<grid, block, 0, stream>>>(
        log_A_real, A_imag, B_re, B_im, C_re, C_im, log_dt, out);
}